// VectorQuantizer_27754078666910
// MI455X (gfx1250) — compile-verified
//
#include <hip/hip_runtime.h>

typedef float v2f __attribute__((ext_vector_type(2)));
typedef float v8f __attribute__((ext_vector_type(8)));

#define DDIM 64
#define KCW  8192
#define HWSZ 1024      // H*W
#define NVEC 8192      // B*H*W
#define EPSR 1e-6f

// ---------------- Kernel 1: codeword squared norms S[k] = sum_d E[d][k]^2 ----
__global__ void __launch_bounds__(256) vq_enorm(const float* __restrict__ E,
                                                float* __restrict__ S) {
    const int k = blockIdx.x * 256 + threadIdx.x;   // coalesced across k
    float s = 0.f;
#pragma unroll
    for (int d = 0; d < DDIM; ++d) {
        float v = E[d * KCW + k];
        s += v * v;
    }
    S[k] = s;
}

// ---------------- Kernel 1b: pack E into WMMA-native B-fragment layout ------
// P[kt*512 + t*32 + lane] = { E[4t+2g][kt*16+m], E[4t+2g+1][kt*16+m] }
// so the hot loop reads 16 coalesced b64 loads per tile (immediate offsets).
__global__ void __launch_bounds__(256) vq_pack(const float* __restrict__ E,
                                               v2f* __restrict__ P) {
    const int id   = blockIdx.x * 256 + threadIdx.x;  // (kt, t, lane) flat
    const int lane = id & 31;
    const int t    = (id >> 5) & 15;
    const int kt   = id >> 9;
    const int m    = lane & 15;
    const int g    = lane >> 4;
    const int col  = kt * 16 + m;
    const int k0   = 4 * t + 2 * g;
    v2f v;
    v[0] = E[(size_t)k0 * KCW + col];
    v[1] = E[(size_t)(k0 + 1) * KCW + col];
    P[id] = v;
}

// ---------------- Kernel 2: fused GEMM(dot) + argmin via WMMA f32 16x16x4 ----
// One wave per 16-row block of flat_x; sweeps 512 tiles of 16 codewords.
// 16 V_WMMA_F32_16X16X4_F32 per tile split over 4 independent acc chains.
__global__ void __launch_bounds__(32) vq_argmin(const float* __restrict__ X,
                                                const v2f* __restrict__ P,
                                                const float* __restrict__ S,
                                                int* __restrict__ IND,
                                                float* __restrict__ INDF) {
    const int lane = threadIdx.x & 31;
    const int rb   = blockIdx.x;          // row block (16 vectors)
    const int n0   = rb * 16;
    const int b    = n0 >> 10;            // batch (blocks never straddle b)
    const int p0   = n0 & 1023;           // h*W + w base
    const int m    = lane & 15;
    const int g    = lane >> 4;

    // A fragments: 16 x-rows, all 64 k-values, resident in VGPRs.
    const float* Xr = X + (size_t)b * DDIM * HWSZ + p0 + m;
    v2f afrag[16];
#pragma unroll
    for (int t = 0; t < 16; ++t) {
        const int k0 = 4 * t + 2 * g;
        afrag[t][0] = Xr[(size_t)k0 * HWSZ];
        afrag[t][1] = Xr[(size_t)(k0 + 1) * HWSZ];
    }

    float minv[8];
    int   mini[8];
#pragma unroll
    for (int j = 0; j < 8; ++j) { minv[j] = 3.4e38f; mini[j] = 0; }

    for (int kt = 0; kt < KCW / 16; ++kt) {
        const int col = kt * 16 + m;            // this lane's codeword column
        const v2f* Pt = P + (size_t)kt * 512 + lane;
        v2f bfrag[16];
#pragma unroll
        for (int t = 0; t < 16; ++t) bfrag[t] = Pt[t * 32];

        // 4 independent accumulation chains to break the WMMA RAW chain
        v8f acc0 = {}, acc1 = {}, acc2 = {}, acc3 = {};
#pragma unroll
        for (int t = 0; t < 16; t += 4) {
            acc0 = __builtin_amdgcn_wmma_f32_16x16x4_f32(
                false, afrag[t + 0], false, bfrag[t + 0], (short)0, acc0, false, false);
            acc1 = __builtin_amdgcn_wmma_f32_16x16x4_f32(
                false, afrag[t + 1], false, bfrag[t + 1], (short)0, acc1, false, false);
            acc2 = __builtin_amdgcn_wmma_f32_16x16x4_f32(
                false, afrag[t + 2], false, bfrag[t + 2], (short)0, acc2, false, false);
            acc3 = __builtin_amdgcn_wmma_f32_16x16x4_f32(
                false, afrag[t + 3], false, bfrag[t + 3], (short)0, acc3, false, false);
        }

        const float sc = S[col];
#pragma unroll
        for (int j = 0; j < 8; ++j) {
            const float dot  = (acc0[j] + acc1[j]) + (acc2[j] + acc3[j]);
            const float dist = sc - 2.0f * dot;     // + ||x||^2 (row const)
            // col strictly increases per lane -> strict < keeps first argmin
            if (dist < minv[j]) { minv[j] = dist; mini[j] = col; }
        }
    }

    // Argmin reduce across the 16 lanes of each half-wave (xor<16 stays in half)
#pragma unroll
    for (int off = 1; off < 16; off <<= 1) {
#pragma unroll
        for (int j = 0; j < 8; ++j) {
            const float ov = __shfl_xor(minv[j], off, 32);
            const int   oi = __shfl_xor(mini[j], off, 32);
            if (ov < minv[j] || (ov == minv[j] && oi < mini[j])) {
                minv[j] = ov;
                mini[j] = oi;
            }
        }
    }
    if (m == 0) {   // lanes 0 and 16 cover rows m = j and m = j + 8
#pragma unroll
        for (int j = 0; j < 8; ++j) {
            const int row = n0 + j + 8 * g;
            IND[row]  = mini[j];
            INDF[row] = (float)mini[j];
        }
    }
}

// ---------------- Kernel 3: gather + rotation trick, one wave per vector ----
__global__ void __launch_bounds__(256) vq_rotate(const float* __restrict__ X,
                                                 const float* __restrict__ E,
                                                 const int* __restrict__ IND,
                                                 float* __restrict__ QD,
                                                 float* __restrict__ Q) {
    const int lane = threadIdx.x & 31;
    const int wave = threadIdx.x >> 5;
    const int n    = blockIdx.x * 8 + wave;
    const int b    = n >> 10;
    const int p    = n & 1023;
    const int idx  = IND[n];

    const float* Xv = X + (size_t)b * DDIM * HWSZ + p;   // elem d at Xv[d*HW]
    const int d0 = lane, d1 = lane + 32;
    const float e0 = Xv[(size_t)d0 * HWSZ];
    const float e1 = Xv[(size_t)d1 * HWSZ];
    const float q0 = E[(size_t)d0 * KCW + idx];
    const float q1 = E[(size_t)d1 * KCW + idx];

    float ee = e0 * e0 + e1 * e1;
    float qq = q0 * q0 + q1 * q1;
#pragma unroll
    for (int off = 1; off < 32; off <<= 1) {
        ee += __shfl_xor(ee, off, 32);
        qq += __shfl_xor(qq, off, 32);
    }
    const float en  = sqrtf(ee) + EPSR;
    const float qn  = sqrtf(qq) + EPSR;
    const float lam = qn / en;
    const float eh0 = e0 / en, eh1 = e1 / en;
    const float qh0 = q0 / qn, qh1 = q1 / qn;
    float r0 = eh0 + qh0, r1 = eh1 + qh1;

    float rr = r0 * r0 + r1 * r1;
    float re = r0 * e0 + r1 * e1;   // (unnormalized r) . e
    float he = eh0 * e0 + eh1 * e1; // e_hat . e
#pragma unroll
    for (int off = 1; off < 32; off <<= 1) {
        rr += __shfl_xor(rr, off, 32);
        re += __shfl_xor(re, off, 32);
        he += __shfl_xor(he, off, 32);
    }
    const float rinv = rsqrtf(rr);
    const float rde  = re * rinv;          // (normalized r) . e

    const float qd0 = lam * (e0 - 2.f * (r0 * rinv) * rde + 2.f * qh0 * he);
    const float qd1 = lam * (e1 - 2.f * (r1 * rinv) * rde + 2.f * qh1 * he);

    float* qdv = QD + (size_t)b * DDIM * HWSZ + p;
    float* qv  = Q  + (size_t)b * DDIM * HWSZ + p;
    qdv[(size_t)d0 * HWSZ] = qd0;
    qdv[(size_t)d1 * HWSZ] = qd1;
    qv[(size_t)d0 * HWSZ]  = q0;
    qv[(size_t)d1 * HWSZ]  = q1;
}

extern "C" void kernel_launch(void* const* d_in, const int* in_sizes, int n_in,
                              void* d_out, int out_size, void* d_ws, size_t ws_size,
                              hipStream_t stream) {
    const float* X = (const float*)d_in[0];   // (B, D, H, W) = (8, 64, 32, 32)
    const float* E = (const float*)d_in[1];   // (D, K) = (64, 8192)

    float* out  = (float*)d_out;
    float* QD   = out;                         // (B, D, H, W)
    float* Q    = out + 8 * 64 * 32 * 32;      // (B, D, H, W)
    float* INDF = out + 2 * 8 * 64 * 32 * 32;  // (B, H, W) as float

    float* S   = (float*)d_ws;                               // K floats (32KB)
    int*   IND = (int*)((char*)d_ws + KCW * sizeof(float));  // N ints  (32KB)
    v2f*   P   = (v2f*)((char*)d_ws + 2 * KCW * sizeof(float)); // 2MB packed E

    vq_enorm<<<KCW / 256, 256, 0, stream>>>(E, S);
    vq_pack<<<(KCW / 16) * 16 * 32 / 256, 256, 0, stream>>>(E, P);
    vq_argmin<<<NVEC / 16, 32, 0, stream>>>(X, P, S, IND, INDF);
    vq_rotate<<<NVEC / 8, 256, 0, stream>>>(X, E, IND, QD, Q);
}